// FourierKANAdapter_42958262895276
// MI455X (gfx1250) — compile-verified
//
#include <hip/hip_runtime.h>

typedef float        v4f   __attribute__((ext_vector_type(4)));
typedef unsigned int u32x4 __attribute__((ext_vector_type(4)));
typedef int          i32x8 __attribute__((ext_vector_type(8)));
typedef int          i32x4 __attribute__((ext_vector_type(4)));

#define DIM            1024
#define NCOEF          7
#define COEF_ELEMS     (DIM * NCOEF)      // 7168 floats = 28 KB
#define ROWS_PER_BLOCK 16
#define BLOCK          256                // 8 wave32 waves

__global__ __launch_bounds__(BLOCK)
void fourier_kan_kernel(const float* __restrict__ x,
                        const float* __restrict__ coeffs,
                        float* __restrict__ out)
{
    __shared__ float scoef[COEF_ELEMS];

    // ---- Stage the 28 KB coefficient table into LDS with the Tensor Data
    // ---- Mover (one DMA issued by wave 0; TENSORcnt tracks completion).
    if (threadIdx.x < 32u) {
        unsigned long long ga = (unsigned long long)(const void*)coeffs;
        unsigned int lds_off  = (unsigned int)(unsigned long long)(void*)&scoef[0];

        u32x4 g0;
        g0[0] = 1u;                                   // count=1, is_restore=0, gather off
        g0[1] = lds_off;                              // lds_addr (bytes)
        g0[2] = (unsigned int)(ga & 0xFFFFFFFFu);     // global_addr[31:0]
        g0[3] = ((unsigned int)(ga >> 32) & 0x01FFFFFFu)
              | (2u << 30);                           // global_addr[56:32] | type=2

        i32x8 g1;
        g1[0] = 0x00020000;                           // workgroup_mask=0, data_size=4B
        g1[1] = (int)((unsigned)(COEF_ELEMS & 0xFFFF) << 16); // tensor_dim0 lo16 @bits[63:48]
        g1[2] = (int)(1u << 16);                      // tensor_dim0 hi16=0, tensor_dim1=1
        g1[3] = (int)((unsigned)(COEF_ELEMS & 0xFFFF) << 16); // tile_dim0=7168
        g1[4] = 0;                                    // tile_dim1/2 unused (1-D tile)
        g1[5] = COEF_ELEMS;                           // tensor_dim0_stride low32
        g1[6] = 0;
        g1[7] = 0;

        i32x4 z4 = {0, 0, 0, 0};                      // groups 2/3 unused (<=2D tensor)
        i32x8 z8 = {0, 0, 0, 0, 0, 0, 0, 0};          // extra group (clang-23 6-arg form)
        __builtin_amdgcn_tensor_load_to_lds(g0, g1, z4, z4, z8, 0);
        __builtin_amdgcn_s_wait_tensorcnt(0);
    }
    __syncthreads();

    // ---- Each thread owns 4 fixed columns; pull its 28 coeffs into VGPRs.
    const int tid = threadIdx.x;
    float cf[28];
    {
        const v4f* s4 = (const v4f*)&scoef[tid * 28]; // 112*tid bytes, 16B aligned
        #pragma unroll
        for (int q = 0; q < 7; ++q) ((v4f*)cf)[q] = s4[q];
    }

    // ---- Stream ROWS_PER_BLOCK rows: NT 128-bit load -> math -> NT store.
    const long long base = (long long)blockIdx.x * ROWS_PER_BLOCK * DIM + tid * 4;
    const float* xp = x   + base;
    float*       op = out + base;

    #pragma unroll 4
    for (int r = 0; r < ROWS_PER_BLOCK; ++r) {
        const long long off = (long long)r * DIM;
        if (r + 1 < ROWS_PER_BLOCK)
            __builtin_prefetch(xp + off + DIM, 0, 0); // global_prefetch_b8, next row

        v4f xv = __builtin_nontemporal_load((const v4f*)(xp + off));
        v4f ov;
        #pragma unroll
        for (int j = 0; j < 4; ++j) {
            const float xx = xv[j];
            // k=1 via HW trans ops, k=2,3 via angle-addition (2 trans/elem total)
            const float s1 = __sinf(xx);
            const float c1 = __cosf(xx);
            const float s2 = 2.0f * s1 * c1;
            const float c2 = 1.0f - 2.0f * s1 * s1;
            const float s3 = s1 * c2 + c1 * s2;
            const float c3 = c1 * c2 - s1 * s2;
            const float* c = &cf[j * 7];              // [c0, s1c, c1c, s2c, c2c, s3c, c3c]
            float corr = c[0];
            corr = fmaf(s1, c[1], corr);
            corr = fmaf(c1, c[2], corr);
            corr = fmaf(s2, c[3], corr);
            corr = fmaf(c2, c[4], corr);
            corr = fmaf(s3, c[5], corr);
            corr = fmaf(c3, c[6], corr);
            ov[j] = xx + corr;
        }
        __builtin_nontemporal_store(ov, (v4f*)(op + off));
    }
}

extern "C" void kernel_launch(void* const* d_in, const int* in_sizes, int n_in,
                              void* d_out, int out_size, void* d_ws, size_t ws_size,
                              hipStream_t stream)
{
    const float* x      = (const float*)d_in[0];
    const float* coeffs = (const float*)d_in[1];
    float*       out    = (float*)d_out;

    const int T    = in_sizes[0] / DIM;        // 32768
    const int grid = T / ROWS_PER_BLOCK;       // 2048 blocks of 256 threads

    fourier_kan_kernel<<<grid, BLOCK, 0, stream>>>(x, coeffs, out);
}